// NEUROPULSBonus_anymatrixNxN_56358560858276
// MI455X (gfx1250) — compile-verified
//
#include <hip/hip_runtime.h>
#include <math.h>

// ---------------------------------------------------------------------------
// MI455X (gfx1250) implementation of the NEUROPULS 128-port MZI mesh.
//
// Strategy (latency-bound problem; see analysis):
//   K1: 8 chunks x 256 identity columns, wave-per-column register-resident
//       VALU chain (2x2 complex mixes, __shfl at lane boundaries).
//   K2: tree-combine the 8 chunk transfer matrices with dense complex f32
//       GEMMs on v_wmma_f32_16x16x4_f32 (7 GEMMs, 3 sequential rounds).
//   K3: elementwise epilogue applying MMI_IN / MMI_OUT / input-output phases.
// ---------------------------------------------------------------------------

typedef __attribute__((ext_vector_type(2))) float v2f;
typedef __attribute__((ext_vector_type(8))) float v8f;

#define NCH        8
#define NOPS       382
#define MAT_ELEMS  (256 * 256)   // float2 elements per 256x256 complex matrix

// ---------------------------------------------------------------------------
// K1: build chunk transfer matrices.  grid = 256 blocks (8 chunks x 32 column
// slices), block = 256 threads = 8 waves, one wave per identity column.
// Each lane holds 8 consecutive complex rows in registers.
// ---------------------------------------------------------------------------
__global__ __launch_bounds__(256) void chunk_kernel(const float* __restrict__ theta_even,
                                                    float2* __restrict__ ws) {
    const int lane  = threadIdx.x & 31;
    const int wave  = threadIdx.x >> 5;
    const int c     = blockIdx.x >> 5;       // chunk id 0..7
    const int slice = blockIdx.x & 31;
    const int col   = slice * 8 + wave;      // identity column 0..255

    // mesh constants (compile-time folded)
    const float eD = sqrtf(0.95f) * sqrtf(0.505f);  // MMI_EVEN diag (real)
    const float eO = sqrtf(0.95f) * sqrtf(0.495f);  // MMI_EVEN offdiag (coeff of i)
    const float cD = sqrtf(0.98f) * sqrtf(0.01f);   // CROSS diag (real)
    const float cO = sqrtf(0.98f) * sqrtf(0.99f);   // CROSS offdiag (coeff of i)
    const float cBar = sqrtf(0.98f) * sqrtf(0.99f); // CROSS corner scalar (real)

    float xr[8], xi[8];
#pragma unroll
    for (int m = 0; m < 8; ++m) {
        const int row = lane * 8 + m;
        xr[m] = (row == col) ? 1.0f : 0.0f;
        xi[m] = 0.0f;
    }

    const int t0 = (c * NOPS) / NCH;
    const int t1 = ((c + 1) * NOPS) / NCH;

    for (int t = t0; t < t1; ++t) {
        // ---- decode op t -> (phase row s, matrix type mt: 0=E,1=C,2=none) ----
        int s, mt;
        if (t == 0)        { s = 0;   mt = 0; }
        else if (t == 1)   { s = -1;  mt = 1; }
        else if (t < 380)  {
            const int j = (t - 2) / 3, r = (t - 2) % 3;
            if (r == 0)      { s = 2 * j + 1; mt = 0; }
            else if (r == 1) { s = 2 * j + 2; mt = 0; }
            else             { s = -1;        mt = 1; }
        }
        else if (t == 380) { s = 253; mt = 0; }
        else               { s = 254; mt = 2; }

        // ---- diagonal phase on even rows ----
        if (s >= 0) {
            const float* th = theta_even + s * 128 + lane * 4;
            // stream upcoming phase rows toward L2 (global_prefetch_b8)
            __builtin_prefetch(theta_even + ((s + 8 < 255) ? (s + 8) : s) * 128 + lane * 4, 0, 1);
#pragma unroll
            for (int m2 = 0; m2 < 4; ++m2) {
                float sn, cs;
                __sincosf(th[m2], &sn, &cs);
                const float r = xr[2 * m2], i = xi[2 * m2];
                xr[2 * m2] = r * cs - i * sn;
                xi[2 * m2] = r * sn + i * cs;
            }
        }

        if (mt == 0) {
            // MMI_EVEN: pairs (2k, 2k+1), all in-lane. y0 = eD*x0 + i*eO*x1, sym.
#pragma unroll
            for (int p = 0; p < 4; ++p) {
                const float ar = xr[2 * p],     ai = xi[2 * p];
                const float br = xr[2 * p + 1], bi = xi[2 * p + 1];
                xr[2 * p]     = eD * ar - eO * bi;
                xi[2 * p]     = eD * ai + eO * br;
                xr[2 * p + 1] = eD * br - eO * ai;
                xi[2 * p + 1] = eD * bi + eO * ar;
            }
        } else if (mt == 1) {
            // CROSS: pairs (2k+1, 2k+2); lane-boundary pair needs 2 shuffles.
            const float xnr = __shfl_down(xr[0], 1, 32);  // next lane's row 8(L+1)
            const float xni = __shfl_down(xi[0], 1, 32);
            const float xpr = __shfl_up(xr[7], 1, 32);    // prev lane's row 8L-1
            const float xpi = __shfl_up(xi[7], 1, 32);
#pragma unroll
            for (int p = 0; p < 3; ++p) {
                const int a = 2 * p + 1, b = 2 * p + 2;
                const float ar = xr[a], ai = xi[a];
                const float br = xr[b], bi = xi[b];
                xr[a] = cD * ar - cO * bi;
                xi[a] = cD * ai + cO * br;
                xr[b] = cD * br - cO * ai;
                xi[b] = cD * bi + cO * ar;
            }
            if (lane < 31) {                 // pair (8L+7, 8L+8), upper half
                const float ar = xr[7], ai = xi[7];
                xr[7] = cD * ar - cO * xni;
                xi[7] = cD * ai + cO * xnr;
            } else {                         // global row 255: corner scalar
                xr[7] *= cBar; xi[7] *= cBar;
            }
            if (lane > 0) {                  // pair (8L-1, 8L), lower half
                const float br = xr[0], bi = xi[0];
                xr[0] = cD * br - cO * xpi;
                xi[0] = cD * bi + cO * xpr;
            } else {                         // global row 0: corner scalar
                xr[0] *= cBar; xi[0] *= cBar;
            }
        }
    }

    // write column `col` of T_c (row-major float2)
    float2* Tc = ws + (size_t)c * MAT_ELEMS;
#pragma unroll
    for (int m = 0; m < 8; ++m) {
        const int row = lane * 8 + m;
        Tc[row * 256 + col] = make_float2(xr[m], xi[m]);
    }
}

// ---------------------------------------------------------------------------
// K2: C = A * B, 256x256 complex f32, on v_wmma_f32_16x16x4_f32.
// grid = 32 blocks x 256 threads = 256 waves; one 16x16 output tile per wave.
// Complex product = 4 real WMMA accumulations per K-slice of 4.
// ---------------------------------------------------------------------------
__global__ __launch_bounds__(256) void cgemm256_kernel(const float2* __restrict__ A,
                                                       const float2* __restrict__ B,
                                                       float2* __restrict__ C) {
    const int lane = threadIdx.x & 31;
    const int wave = threadIdx.x >> 5;
    const int tid  = blockIdx.x * 8 + wave;   // tile id 0..255
    const int tm   = tid >> 4, tn = tid & 15;
    const int half = lane >> 4, lm = lane & 15;
    const int mrow = tm * 16 + lm;            // A / C row (M dim)
    const int ncol = tn * 16 + lm;            // B / C col (N dim)

    v8f cr = {0.f, 0.f, 0.f, 0.f, 0.f, 0.f, 0.f, 0.f};
    v8f ci = {0.f, 0.f, 0.f, 0.f, 0.f, 0.f, 0.f, 0.f};

    for (int kk = 0; kk < 64; ++kk) {
        const int kb = 4 * kk + 2 * half;     // A layout: VGPR0=K{0,2}, VGPR1=K{1,3}
        // A fragment: two consecutive complex elems -> one 16B load (aligned)
        const float4 af = *reinterpret_cast<const float4*>(A + (size_t)mrow * 256 + kb);
        const v2f ar = {af.x, af.z};
        const v2f ai = {af.y, af.w};
        // B fragment (mirror layout: half-wave splits K)
        const float2 b0 = B[(size_t)kb * 256 + ncol];
        const float2 b1 = B[(size_t)(kb + 1) * 256 + ncol];
        const v2f br  = { b0.x,  b1.x};
        const v2f bi  = { b0.y,  b1.y};
        const v2f nbi = {-b0.y, -b1.y};       // F32 WMMA NEG bits are C-only (ISA 7.12)

        cr = __builtin_amdgcn_wmma_f32_16x16x4_f32(false, ar, false, br,  (short)0, cr, false, false);
        cr = __builtin_amdgcn_wmma_f32_16x16x4_f32(false, ai, false, nbi, (short)0, cr, false, false);
        ci = __builtin_amdgcn_wmma_f32_16x16x4_f32(false, ar, false, bi,  (short)0, ci, false, false);
        ci = __builtin_amdgcn_wmma_f32_16x16x4_f32(false, ai, false, br,  (short)0, ci, false, false);
    }

    // C/D layout: VGPR v holds M=v (lanes 0-15) and M=v+8 (lanes 16-31)
#pragma unroll
    for (int v = 0; v < 8; ++v) {
        const int row = tm * 16 + v + 8 * half;
        C[(size_t)row * 256 + ncol] = make_float2(cr[v], ci[v]);
    }
}

// ---------------------------------------------------------------------------
// K3: out = diag(d_out) * MMI_OUT * T * MMI_IN * diag(d_in)  (128x128 cplx)
// ---------------------------------------------------------------------------
__global__ __launch_bounds__(256) void epilogue_kernel(const float* __restrict__ theta_in,
                                                       const float* __restrict__ theta_out,
                                                       const float2* __restrict__ T,
                                                       float* __restrict__ out) {
    const int id = blockIdx.x * blockDim.x + threadIdx.x;
    if (id >= 128 * 128) return;
    const int i = id >> 7, j = id & 127;

    const float pA = sqrtf(0.95f) * sqrtf(0.505f);  // real weight of v_in/w_out
    const float qA = sqrtf(0.95f) * sqrtf(0.495f);  // imaginary weight (coeff of i)

    const float2 t00 = T[(2 * i)     * 256 + 2 * j];
    const float2 t01 = T[(2 * i)     * 256 + 2 * j + 1];
    const float2 t10 = T[(2 * i + 1) * 256 + 2 * j];
    const float2 t11 = T[(2 * i + 1) * 256 + 2 * j + 1];

    // u_k = T[k][2j]*pA + T[k][2j+1]*(i*qA)
    const float u0r = pA * t00.x - qA * t01.y;
    const float u0i = pA * t00.y + qA * t01.x;
    const float u1r = pA * t10.x - qA * t11.y;
    const float u1i = pA * t10.y + qA * t11.x;
    // z = pA*u0 + (i*qA)*u1
    float zr = pA * u0r - qA * u1i;
    float zi = pA * u0i + qA * u1r;

    float si, cin, so, co;
    __sincosf(theta_in[j], &si, &cin);
    __sincosf(theta_out[i], &so, &co);
    float zr2 = zr * cin - zi * si;
    float zi2 = zr * si + zi * cin;
    float zr3 = zr2 * co - zi2 * so;
    float zi3 = zr2 * so + zi2 * co;

    out[2 * id]     = zr3;   // complex64 interleaved, row-major (i, j)
    out[2 * id + 1] = zi3;
}

// ---------------------------------------------------------------------------
// launch: K1, 7x K2 (tree rounds sequential on the stream), K3.
// Workspace: slots 0..7 chunks, 8..11 round1, 12..13 round2, 14 final T.
// Total 15 * 512KB = 7.5 MB, fully L2-resident (192 MB).
// ---------------------------------------------------------------------------
extern "C" void kernel_launch(void* const* d_in, const int* in_sizes, int n_in,
                              void* d_out, int out_size, void* d_ws, size_t ws_size,
                              hipStream_t stream) {
    (void)in_sizes; (void)n_in; (void)out_size; (void)ws_size;
    const float* theta_in   = (const float*)d_in[0];
    const float* theta_even = (const float*)d_in[1];
    const float* theta_out  = (const float*)d_in[2];
    float2* ws = (float2*)d_ws;

    chunk_kernel<<<256, 256, 0, stream>>>(theta_even, ws);

    auto M = [&](int slot) { return ws + (size_t)slot * MAT_ELEMS; };

    // T_total = T7*T6*...*T0  (later chunks multiply on the left)
    for (int i = 0; i < 4; ++i)   // U_i = T_{2i+1} * T_{2i}
        cgemm256_kernel<<<32, 256, 0, stream>>>(M(2 * i + 1), M(2 * i), M(8 + i));
    for (int i = 0; i < 2; ++i)   // V_i = U_{2i+1} * U_{2i}
        cgemm256_kernel<<<32, 256, 0, stream>>>(M(8 + 2 * i + 1), M(8 + 2 * i), M(12 + i));
    cgemm256_kernel<<<32, 256, 0, stream>>>(M(13), M(12), M(14));  // T = V1 * V0

    epilogue_kernel<<<64, 256, 0, stream>>>(theta_in, theta_out, M(14), (float*)d_out);
}